// PPAttention_35579509080082
// MI455X (gfx1250) — compile-verified
//
#include <hip/hip_runtime.h>
#include <hip/hip_bf16.h>

typedef __attribute__((ext_vector_type(2))) float v2f;
typedef __attribute__((ext_vector_type(4))) float v4f;
typedef __attribute__((ext_vector_type(8))) float v8f;

#define B_ 32
#define S_ 4
#define HID_ 4096
#define H_ 32
#define HKV_ 8
#define D_ 128
#define NPART 8
#define TPART 512   // 4096 / NPART
#define NTILES (TPART / 16)

#if defined(__has_builtin)
#if __has_builtin(__builtin_amdgcn_global_load_async_to_lds_b128)
#define ASYNC_STAGE 1
#endif
#endif
#ifndef ASYNC_STAGE
#define ASYNC_STAGE 0
#endif

#if ASYNC_STAGE
// exact pointer types the builtin expects: int4 in AS1 (global) / AS3 (LDS)
typedef int v4i_ __attribute__((__vector_size__(16)));
typedef __attribute__((address_space(1))) v4i_ as1_v4i;
typedef __attribute__((address_space(3))) v4i_ as3_v4i;
// wait for ALL async LDS loads / for all but the most recent 32 (one tile)
#define WAIT_ASYNC_ALL()  asm volatile("s_wait_asynccnt 0x0" ::: "memory")
#define WAIT_ASYNC_TILE() asm volatile("s_wait_asynccnt 0x20" ::: "memory")
#else
#define WAIT_ASYNC_ALL()  __syncthreads()
#define WAIT_ASYNC_TILE() __syncthreads()
#endif

// Copy 16 bytes (4 floats) per lane into LDS: async DMA path on CDNA5.
__device__ __forceinline__ void stage16(const float* __restrict__ src, float* dst)
{
#if ASYNC_STAGE
  __builtin_amdgcn_global_load_async_to_lds_b128(
      (as1_v4i*)(src), (as3_v4i*)(dst), 0, 0);
#else
  *(v4f*)dst = *(const v4f*)src;
#endif
}

// ---------------------------------------------------------------------------
// Generic fp32 GEMM via V_WMMA_F32_16X16X4_F32, compile-time N/K so the
// B-operand addressing strength-reduces to immediate offsets.
// C[M,N] = A[M,K] * B[K,N], row-major. One wave computes a 64x16 tile.
// grid = (M/64, N/16), block = 32 (one wave32).
// ---------------------------------------------------------------------------
template <int N, int K>
__global__ __launch_bounds__(32) void gemm_f32_wmma(
    const float* __restrict__ A, const float* __restrict__ Bm,
    float* __restrict__ C)
{
  const int lane = threadIdx.x & 31;
  const int half = lane >> 4;      // 0: lanes 0-15, 1: lanes 16-31
  const int lcol = lane & 15;
  const int koff = half * 2;       // A/B fragment K offset per lane half
  const int m0 = blockIdx.x * 64;
  const int n0 = blockIdx.y * 16;

  v8f c0 = {}, c1 = {}, c2 = {}, c3 = {};

  const float* Ar0 = A + (size_t)(m0 + 0  + lcol) * K + koff;
  const float* Ar1 = A + (size_t)(m0 + 16 + lcol) * K + koff;
  const float* Ar2 = A + (size_t)(m0 + 32 + lcol) * K + koff;
  const float* Ar3 = A + (size_t)(m0 + 48 + lcol) * K + koff;
  const float* Bp  = Bm + (size_t)koff * N + n0 + lcol;

#pragma unroll 8
  for (int kk = 0; kk < K; kk += 4) {
    v2f bf;
    bf.x = Bp[(size_t)kk * N];
    bf.y = Bp[(size_t)kk * N + N];
    v2f a0 = *(const v2f*)(Ar0 + kk);
    v2f a1 = *(const v2f*)(Ar1 + kk);
    v2f a2 = *(const v2f*)(Ar2 + kk);
    v2f a3 = *(const v2f*)(Ar3 + kk);
    c0 = __builtin_amdgcn_wmma_f32_16x16x4_f32(false, a0, false, bf, (short)0, c0, false, false);
    c1 = __builtin_amdgcn_wmma_f32_16x16x4_f32(false, a1, false, bf, (short)0, c1, false, false);
    c2 = __builtin_amdgcn_wmma_f32_16x16x4_f32(false, a2, false, bf, (short)0, c2, false, false);
    c3 = __builtin_amdgcn_wmma_f32_16x16x4_f32(false, a3, false, bf, (short)0, c3, false, false);
  }

  // C layout: VGPR j -> row (j + 8*half), col = lcol
  float* Cp = C + n0 + lcol;
#pragma unroll
  for (int j = 0; j < 8; ++j) {
    const int r = m0 + j + 8 * half;
    Cp[(size_t)(r +  0) * N] = c0[j];
    Cp[(size_t)(r + 16) * N] = c1[j];
    Cp[(size_t)(r + 32) * N] = c2[j];
    Cp[(size_t)(r + 48) * N] = c3[j];
  }
}

// ---------------------------------------------------------------------------
// RoPE Q + repack into per-(b,kv) 16x128 tiles (rows = g*4 + s), prescaled
// by 1/sqrt(D). Exactly B*S*H*D threads.
// ---------------------------------------------------------------------------
__global__ void rope_q_kernel(const float* __restrict__ rawQ,
                              const float* __restrict__ cosT,
                              const float* __restrict__ sinT,
                              const int* __restrict__ past_ptr,
                              float* __restrict__ Qr)
{
  const int tid = blockIdx.x * blockDim.x + threadIdx.x;
  const int d = tid & 127;
  const int h = (tid >> 7) & 31;
  const int s = (tid >> 12) & 3;
  const int b = tid >> 14;
  const int past = *past_ptr;
  const int pos = past + s;
  const float cv = cosT[pos * D_ + d];
  const float sv = sinT[pos * D_ + d];
  const float* q = rawQ + ((size_t)((b * S_ + s) * H_ + h) << 7);
  const float t = q[d];
  const float rot = (d < 64) ? -q[d + 64] : q[d - 64];
  const float val = (t * cv + rot * sv) * 0.08838834764831845f; // 1/sqrt(128)
  const int kv = h >> 2, g = h & 3;
  Qr[((size_t)((b * HKV_ + kv) * 16 + (g * 4 + s)) << 7) + d] = val;
}

// RoPE K + pack, and pack V, into (b, kv, s, d). Exactly B*S*HKV*D threads.
__global__ void rope_kv_kernel(const float* __restrict__ rawK,
                               const float* __restrict__ rawV,
                               const float* __restrict__ cosT,
                               const float* __restrict__ sinT,
                               const int* __restrict__ past_ptr,
                               float* __restrict__ k_new,
                               float* __restrict__ v_new)
{
  const int tid = blockIdx.x * blockDim.x + threadIdx.x;
  const int d = tid & 127;
  const int kv = (tid >> 7) & 7;
  const int s = (tid >> 10) & 3;
  const int b = tid >> 12;
  const int past = *past_ptr;
  const int pos = past + s;
  const float cv = cosT[pos * D_ + d];
  const float sv = sinT[pos * D_ + d];
  const size_t src = (size_t)((b * S_ + s) * HKV_ + kv) << 7;
  const float* kp = rawK + src;
  const float t = kp[d];
  const float rot = (d < 64) ? -kp[d + 64] : kp[d - 64];
  const size_t dst = ((size_t)((b * HKV_ + kv) * S_ + s) << 7) + d;
  k_new[dst] = t * cv + rot * sv;
  v_new[dst] = rawV[src + d];
}

// ---------------------------------------------------------------------------
// Flash-decode attention partition. One wave per (b*HKV+kv, partition).
// 16 query rows = (g,s) pairs. Streams TPART t-values in tiles of 16.
// K/V tiles double-buffered in LDS via async global->LDS DMA; scores and
// P*V both via WMMA f32 16x16x4.
// ---------------------------------------------------------------------------
__global__ __launch_bounds__(32) void attn_flash(
    const float* __restrict__ Qr,
    const float* __restrict__ k_cache, const float* __restrict__ v_cache,
    const float* __restrict__ k_new,   const float* __restrict__ v_new,
    const int* __restrict__ past_ptr,
    float* __restrict__ Opart, float* __restrict__ mpart, float* __restrict__ lpart)
{
  __shared__ __align__(16) float ks[2][16][136]; // 128 + pad 8 (16B rows, bank spread)
  __shared__ __align__(16) float vs[2][16][136];
  __shared__ __align__(16) float ps[16][18];     // P tile, pad to keep v2f aligned

  const int lane = threadIdx.x;
  const int half = lane >> 4;
  const int lcol = lane & 15;
  const int koff = half * 2;
  const int bkv = blockIdx.x;            // b*HKV + kv
  const int part = blockIdx.y;
  const int b = bkv >> 3, kv = bkv & 7;
  const int past = *past_ptr;

  // A-fragments of Q (16x128): 32 fragments of K=4 each.
  const float* qbase = Qr + ((size_t)bkv * 16) * D_ + lcol * D_;
  v2f qa[32];
#pragma unroll
  for (int i = 0; i < 32; ++i)
    qa[i] = *(const v2f*)(qbase + 4 * i + koff);

  v8f o[8];
  float mrow[8], lrow[8];
#pragma unroll
  for (int j = 0; j < 8; ++j) { o[j] = (v8f){}; mrow[j] = -3.0e38f; lrow[j] = 0.0f; }

  // issue the 32 async (or sync) row copies for one 16-t tile into buffer buf
  auto stage_tile = [&](int t0, int buf) {
    for (int r = 0; r < 16; ++r) {
      const int t = t0 + r;
      const float *ksrc, *vsrc;
      if (t < past) {
        const size_t off = (((size_t)b * past + t) * HKV_ + kv) * D_;
        ksrc = k_cache + off;
        vsrc = v_cache + off;
      } else {
        const size_t off = ((size_t)bkv * S_ + (t - past)) * D_;
        ksrc = k_new + off;
        vsrc = v_new + off;
      }
      stage16(ksrc + lane * 4, &ks[buf][r][lane * 4]);
      stage16(vsrc + lane * 4, &vs[buf][r][lane * 4]);
    }
  };

  const int tbase = part * TPART;
  stage_tile(tbase, 0);

  for (int tt = 0; tt < NTILES; ++tt) {
    const int t0 = tbase + tt * 16;
    const int buf = tt & 1;

    // prefetch next tile into the other buffer, then wait for this one
    if (tt + 1 < NTILES) {
      stage_tile(tbase + (tt + 1) * 16, buf ^ 1);
      WAIT_ASYNC_TILE();
    } else {
      WAIT_ASYNC_ALL();
    }

    // scores tile (16 rows x 16 t): C += Q(16x128) * K^T(128x16)
    v8f c = {};
#pragma unroll
    for (int i = 0; i < 32; ++i) {
      v2f bk = *(const v2f*)(&ks[buf][lcol][4 * i + koff]);
      c = __builtin_amdgcn_wmma_f32_16x16x4_f32(false, qa[i], false, bk, (short)0, c, false, false);
    }

    // causal mask (only boundary tile): valid iff t <= past + s, s = row & 3
    if (t0 + 15 > past) {
      const int t = t0 + lcol;
#pragma unroll
      for (int j = 0; j < 8; ++j)
        if (t > past + (j & 3)) c[j] = -3.0e38f;
    }

    // online softmax: per-row max/sum via cross-lane xor reductions (16 lanes)
    float alpha[8];
#pragma unroll
    for (int j = 0; j < 8; ++j) {
      float tm = c[j];
      tm = fmaxf(tm, __shfl_xor(tm, 1));
      tm = fmaxf(tm, __shfl_xor(tm, 2));
      tm = fmaxf(tm, __shfl_xor(tm, 4));
      tm = fmaxf(tm, __shfl_xor(tm, 8));
      const float mn = fmaxf(mrow[j], tm);
      const float al = __expf(mrow[j] - mn);
      const float pj = __expf(c[j] - mn);
      float rs = pj;
      rs += __shfl_xor(rs, 1);
      rs += __shfl_xor(rs, 2);
      rs += __shfl_xor(rs, 4);
      rs += __shfl_xor(rs, 8);
      lrow[j] = lrow[j] * al + rs;
      mrow[j] = mn;
      alpha[j] = al;
      ps[j + 8 * half][lcol] = pj;   // stage P for transpose to A-layout
    }
#pragma unroll
    for (int j8 = 0; j8 < 8; ++j8)
#pragma unroll
      for (int j = 0; j < 8; ++j)
        o[j8][j] *= alpha[j];

    // O += P(16x16) * V(16x128)   (per-wave LDS ops are in-order: the ps
    // stores above are seen by the ds reads below)
#pragma unroll
    for (int i = 0; i < 4; ++i) {
      v2f ap = *(const v2f*)(&ps[lcol][4 * i + koff]);
#pragma unroll
      for (int j8 = 0; j8 < 8; ++j8) {
        v2f bv;
        bv.x = vs[buf][4 * i + koff][j8 * 16 + lcol];
        bv.y = vs[buf][4 * i + koff + 1][j8 * 16 + lcol];
        o[j8] = __builtin_amdgcn_wmma_f32_16x16x4_f32(false, ap, false, bv, (short)0, o[j8], false, false);
      }
    }

    // drain pending LDS reads before the async engine may overwrite this buf
    asm volatile("s_wait_dscnt 0x0" ::: "memory");
  }

  // write partials
  const size_t obase = ((size_t)bkv * NPART + part) * 16 * (size_t)D_;
#pragma unroll
  for (int j8 = 0; j8 < 8; ++j8)
#pragma unroll
    for (int j = 0; j < 8; ++j) {
      const int r = j + 8 * half;
      Opart[obase + (size_t)r * D_ + j8 * 16 + lcol] = o[j8][j];
    }
  if (lcol == 0) {
    const size_t sbase = ((size_t)bkv * NPART + part) * 16;
#pragma unroll
    for (int j = 0; j < 8; ++j) {
      mpart[sbase + j + 8 * half] = mrow[j];
      lpart[sbase + j + 8 * half] = lrow[j];
    }
  }
}

// ---------------------------------------------------------------------------
// Combine partition partials (log-sum-exp merge) -> attn_out (B,S,H*D).
// grid = B*HKV blocks of 128 threads (one per d).
// ---------------------------------------------------------------------------
__global__ __launch_bounds__(128) void attn_combine(
    const float* __restrict__ Opart, const float* __restrict__ mpart,
    const float* __restrict__ lpart, float* __restrict__ attn_out)
{
  __shared__ float Ws[16][NPART];
  __shared__ float Ls[16];
  const int bkv = blockIdx.x;
  const int b = bkv >> 3, kv = bkv & 7;
  const int tid = threadIdx.x;

  if (tid < 16) {
    const int r = tid;
    const size_t sbase = (size_t)bkv * NPART * 16 + r;
    float M = -3.0e38f;
    for (int p = 0; p < NPART; ++p) M = fmaxf(M, mpart[sbase + p * 16]);
    float L = 0.0f;
    for (int p = 0; p < NPART; ++p) {
      const float w = __expf(mpart[sbase + p * 16] - M);
      Ws[r][p] = w;
      L += lpart[sbase + p * 16] * w;
    }
    Ls[r] = L;
  }
  __syncthreads();

  const int d = tid;
  for (int r = 0; r < 16; ++r) {
    float acc = 0.0f;
    const size_t ob = ((size_t)bkv * NPART * 16 + r) * D_ + d;
    for (int p = 0; p < NPART; ++p)
      acc += Opart[ob + (size_t)p * 16 * D_] * Ws[r][p];
    const int g = r >> 2, s = r & 3;
    const int h = kv * 4 + g;
    attn_out[((size_t)(b * S_ + s) * H_ + h) * D_ + d] = acc / Ls[r];
  }
}

// ---------------------------------------------------------------------------
extern "C" void kernel_launch(void* const* d_in, const int* in_sizes, int n_in,
                              void* d_out, int out_size, void* d_ws, size_t ws_size,
                              hipStream_t stream) {
  (void)in_sizes; (void)n_in; (void)out_size; (void)ws_size;
  const float* x       = (const float*)d_in[0];
  const float* cosT    = (const float*)d_in[1];
  const float* sinT    = (const float*)d_in[2];
  const float* k_cache = (const float*)d_in[3];
  const float* v_cache = (const float*)d_in[4];
  const float* Wq      = (const float*)d_in[5];
  const float* Wk      = (const float*)d_in[6];
  const float* Wv      = (const float*)d_in[7];
  const float* Wo      = (const float*)d_in[8];
  const int*   past_p  = (const int*)d_in[9];
  float* out = (float*)d_out;

  float* ws = (float*)d_ws;
  float* rawQ     = ws; ws += (size_t)B_ * S_ * H_ * D_;        // 524288
  float* rawK     = ws; ws += (size_t)B_ * S_ * HKV_ * D_;      // 131072
  float* rawV     = ws; ws += (size_t)B_ * S_ * HKV_ * D_;
  float* Qr       = ws; ws += (size_t)B_ * HKV_ * 16 * D_;      // 524288
  float* k_new    = ws; ws += (size_t)B_ * HKV_ * S_ * D_;
  float* v_new    = ws; ws += (size_t)B_ * HKV_ * S_ * D_;
  float* attn_out = ws; ws += (size_t)B_ * S_ * H_ * D_;
  float* Opart    = ws; ws += (size_t)B_ * HKV_ * NPART * 16 * D_; // 4194304
  float* mpart    = ws; ws += (size_t)B_ * HKV_ * NPART * 16;
  float* lpart    = ws; ws += (size_t)B_ * HKV_ * NPART * 16;

  const dim3 wave(32);
  // QKV projections (M = B*S = 128)
  gemm_f32_wmma<HID_, HID_><<<dim3(2, HID_ / 16), wave, 0, stream>>>(x, Wq, rawQ);
  gemm_f32_wmma<HKV_ * D_, HID_><<<dim3(2, (HKV_ * D_) / 16), wave, 0, stream>>>(x, Wk, rawK);
  gemm_f32_wmma<HKV_ * D_, HID_><<<dim3(2, (HKV_ * D_) / 16), wave, 0, stream>>>(x, Wv, rawV);
  // RoPE + repack
  rope_q_kernel<<<(B_ * S_ * H_ * D_) / 256, 256, 0, stream>>>(rawQ, cosT, sinT, past_p, Qr);
  rope_kv_kernel<<<(B_ * S_ * HKV_ * D_) / 256, 256, 0, stream>>>(rawK, rawV, cosT, sinT, past_p, k_new, v_new);
  // Flash-decode attention + combine
  attn_flash<<<dim3(B_ * HKV_, NPART), wave, 0, stream>>>(Qr, k_cache, v_cache, k_new, v_new,
                                                          past_p, Opart, mpart, lpart);
  attn_combine<<<B_ * HKV_, 128, 0, stream>>>(Opart, mpart, lpart, attn_out);
  // Output projection
  gemm_f32_wmma<HID_, H_ * D_><<<dim3(2, HID_ / 16), wave, 0, stream>>>(attn_out, Wo, out);
}